// GCNBlock_11338713662112
// MI455X (gfx1250) — compile-verified
//
#include <hip/hip_runtime.h>

typedef __attribute__((ext_vector_type(2))) float v2f;
typedef __attribute__((ext_vector_type(8))) float v8f;

#define C_DIM 128
#define BN_EPS 1e-5f

// ---------------------------------------------------------------------------
// Init: zero [agg | stats], set deg = 1.0 (self-loop weight). One contiguous
// region starting at `base`: first n_zero floats -> 0, next n_one floats -> 1.
// ---------------------------------------------------------------------------
__global__ void init_ws_kernel(float* __restrict__ base, long long n_zero,
                               long long n_one) {
  long long i = (long long)blockIdx.x * blockDim.x + threadIdx.x;
  long long total = n_zero + n_one;
  if (i < total) base[i] = (i < n_zero) ? 0.0f : 1.0f;
}

// ---------------------------------------------------------------------------
// deg[col] += edge_weight  (self-loop "1" already in init)
// ---------------------------------------------------------------------------
__global__ void deg_kernel(const long long* __restrict__ ei,
                           const float* __restrict__ ew,
                           float* __restrict__ deg, int E) {
  int e = blockIdx.x * blockDim.x + threadIdx.x;
  if (e < E) {
    int col = (int)ei[(long long)E + e];
    atomicAdd(&deg[col], ew[e]);
  }
}

// deg -> dinv = deg > 0 ? rsqrt(deg) : 0  (in place)
__global__ void dinv_kernel(float* __restrict__ deg, int n) {
  int i = blockIdx.x * blockDim.x + threadIdx.x;
  if (i < n) {
    float d = deg[i];
    deg[i] = (d > 0.0f) ? rsqrtf(d) : 0.0f;
  }
}

// ---------------------------------------------------------------------------
// H = X @ W using V_WMMA_F32_16X16X4_F32 (fp32 WMMA, keeps BN-grade accuracy).
// Block = 256 threads = 8 waves; block owns one 16-row tile of X, wave w owns
// output columns [16w, 16w+16). K loop: 32 steps of K=4.
//
// f32 VGPR layouts (ISA 7.12.2):
//   A 16x4 : m = lane%16, K = 2*(lane/16) + v   (v = element of v2f)
//   B 4x16 : n = lane%16, K = 2*(lane/16) + v
//   D 16x16: n = lane%16, m = v + 8*(lane/16)   (v = element of v8f)
// ---------------------------------------------------------------------------
__global__ __launch_bounds__(256) void gemm_wmma_kernel(
    const float* __restrict__ X, const float* __restrict__ W,
    float* __restrict__ H, int n) {
  const int wave = threadIdx.x >> 5;
  const int lane = threadIdx.x & 31;
  const int lr = lane & 15;   // row/col within tile
  const int lh = lane >> 4;   // which half of the wave
  const int m0 = blockIdx.x * 16;
  const int n0 = wave * 16;
  if (m0 >= n) return;

  const float* __restrict__ xrow = X + (long long)(m0 + lr) * C_DIM;
  v8f acc = {};
#pragma unroll
  for (int k0 = 0; k0 < C_DIM; k0 += 4) {
    const int ka = k0 + 2 * lh;
    v2f a, b;
    a.x = xrow[ka];
    a.y = xrow[ka + 1];
    b.x = W[(long long)ka * C_DIM + n0 + lr];
    b.y = W[(long long)(ka + 1) * C_DIM + n0 + lr];
    acc = __builtin_amdgcn_wmma_f32_16x16x4_f32(false, a, false, b, (short)0,
                                                acc, false, false);
  }
#pragma unroll
  for (int v = 0; v < 8; ++v) {
    const int m = m0 + v + 8 * lh;
    H[(long long)m * C_DIM + n0 + lr] = acc[v];
  }
}

// ---------------------------------------------------------------------------
// Edge scatter: one wave32 per edge. Lane handles 4 channels (float4 of H row),
// atomic-adds into agg[col]. All traffic stays in L2 (h + agg = ~102MB < 192MB).
// ---------------------------------------------------------------------------
__global__ __launch_bounds__(256) void scatter_kernel(
    const long long* __restrict__ ei, const float* __restrict__ ew,
    const float* __restrict__ dinv, const float* __restrict__ H,
    float* __restrict__ agg, int E) {
  const int wave = blockIdx.x * (blockDim.x >> 5) + (threadIdx.x >> 5);
  const int lane = threadIdx.x & 31;
  if (wave >= E) return;
  const int row = (int)ei[wave];
  const int col = (int)ei[(long long)E + wave];
  const float norm = dinv[row] * ew[wave] * dinv[col];
  const float4 hv =
      *reinterpret_cast<const float4*>(H + (long long)row * C_DIM + lane * 4);
  float* dst = agg + (long long)col * C_DIM + lane * 4;
  atomicAdd(dst + 0, hv.x * norm);
  atomicAdd(dst + 1, hv.y * norm);
  atomicAdd(dst + 2, hv.z * norm);
  atomicAdd(dst + 3, hv.w * norm);
}

// ---------------------------------------------------------------------------
// BN stats: thread = channel, block strides over rows. Self-loop contribution
// h[r][c] * dinv[r]^2 folded in here (saves a full RMW pass over agg).
// ---------------------------------------------------------------------------
__global__ __launch_bounds__(C_DIM) void stats_kernel(
    const float* __restrict__ agg, const float* __restrict__ H,
    const float* __restrict__ dinv, float* __restrict__ stats, int n) {
  const int c = threadIdx.x;
  float s = 0.0f, ss = 0.0f;
  for (int r = blockIdx.x; r < n; r += gridDim.x) {
    const float di = dinv[r];
    const float v =
        agg[(long long)r * C_DIM + c] + H[(long long)r * C_DIM + c] * di * di;
    s += v;
    ss += v * v;
  }
  atomicAdd(&stats[c], s);
  atomicAdd(&stats[C_DIM + c], ss);
}

// ---------------------------------------------------------------------------
// Finalize: y = relu((v - mean) * rsqrt(var + eps) * gamma + beta)
// ---------------------------------------------------------------------------
__global__ __launch_bounds__(256) void final_kernel(
    const float* __restrict__ agg, const float* __restrict__ H,
    const float* __restrict__ dinv, const float* __restrict__ stats,
    const float* __restrict__ gamma, const float* __restrict__ beta,
    float* __restrict__ out, int n) {
  const long long idx = (long long)blockIdx.x * blockDim.x + threadIdx.x;
  const long long total = (long long)n * C_DIM;
  if (idx >= total) return;
  const int c = (int)(idx & (C_DIM - 1));
  const int r = (int)(idx >> 7);
  const float inv_n = 1.0f / (float)n;
  const float mean = stats[c] * inv_n;
  const float var = stats[C_DIM + c] * inv_n - mean * mean;
  const float di = dinv[r];
  const float v = agg[idx] + H[idx] * di * di;
  const float y = (v - mean) * rsqrtf(var + BN_EPS) * gamma[c] + beta[c];
  out[idx] = fmaxf(y, 0.0f);
}

// ---------------------------------------------------------------------------
// Launch
// ---------------------------------------------------------------------------
extern "C" void kernel_launch(void* const* d_in, const int* in_sizes, int n_in,
                              void* d_out, int out_size, void* d_ws,
                              size_t ws_size, hipStream_t stream) {
  const float* x = (const float*)d_in[0];
  const long long* ei = (const long long*)d_in[1];  // int64 edge_index [2,E]
  const float* ew = (const float*)d_in[2];
  const float* W = (const float*)d_in[3];
  const float* gamma = (const float*)d_in[4];
  const float* beta = (const float*)d_in[5];

  const int n = in_sizes[0] / C_DIM;  // 100000
  const int E = in_sizes[1] / 2;      // 1600000

  // Workspace layout (floats): H | agg | stats(2*C) | deg->dinv (N)
  float* ws = (float*)d_ws;
  float* H = ws;
  float* agg = H + (long long)n * C_DIM;
  float* stats = agg + (long long)n * C_DIM;
  float* deg = stats + 2 * C_DIM;  // overwritten with dinv

  const int tpb = 256;
  const long long n_zero = (long long)n * C_DIM + 2 * C_DIM;  // agg + stats
  const long long n_one = n;                                  // deg = 1.0
  const long long init_total = n_zero + n_one;

  init_ws_kernel<<<(int)((init_total + tpb - 1) / tpb), tpb, 0, stream>>>(
      agg, n_zero, n_one);
  deg_kernel<<<(E + tpb - 1) / tpb, tpb, 0, stream>>>(ei, ew, deg, E);
  dinv_kernel<<<(n + tpb - 1) / tpb, tpb, 0, stream>>>(deg, n);
  gemm_wmma_kernel<<<(n + 15) / 16, 256, 0, stream>>>(x, W, H, n);
  scatter_kernel<<<(E + 7) / 8, 256, 0, stream>>>(ei, ew, deg, H, agg, E);
  stats_kernel<<<2048, C_DIM, 0, stream>>>(agg, H, deg, stats, n);
  final_kernel<<<(int)(((long long)n * C_DIM + tpb - 1) / tpb), tpb, 0,
                stream>>>(agg, H, deg, stats, gamma, beta, (float*)d_out, n);
}